// GAT_83640193122824
// MI455X (gfx1250) — compile-verified
//
#include <hip/hip_runtime.h>
#include <math.h>

typedef __attribute__((ext_vector_type(2))) float v2f;
typedef __attribute__((ext_vector_type(8))) float v8f;

#define F_DIM 64
#define H_DIM 4
#define HF    256          // H*F
#define NEG_SLOPE 0.2f
#define LN_EPS 1e-5f

// ---------------------------------------------------------------------------
// float atomic max via sign-aware integer punning (works with init = -inf)
// ---------------------------------------------------------------------------
__device__ __forceinline__ void atomic_max_f32(float* addr, float val) {
    if (val >= 0.0f) atomicMax((int*)addr,          __float_as_int(val));
    else             atomicMin((unsigned int*)addr, __float_as_uint(val));
}

// ---------------------------------------------------------------------------
// 0) init: rst = 0, z = 0, m = -inf
// ---------------------------------------------------------------------------
__global__ void init_kernel(float* __restrict__ rst, float* __restrict__ z,
                            float* __restrict__ m, int nrst, int nz) {
    int i = blockIdx.x * blockDim.x + threadIdx.x;
    if (i < nrst) rst[i] = 0.0f;
    if (i < nz)   { z[i] = 0.0f; m[i] = -INFINITY; }
}

// ---------------------------------------------------------------------------
// 1) feat[N,256] = h[N,64] @ fc_w[64,256]   (exact f32 WMMA 16x16x4)
//    one wave = one 16x16 output tile, K looped in steps of 4
// ---------------------------------------------------------------------------
__global__ void fc_gemm_kernel(const float* __restrict__ h,
                               const float* __restrict__ w,
                               float* __restrict__ feat, int nRowTiles) {
    const int wave   = (blockIdx.x * blockDim.x + threadIdx.x) >> 5;
    const int nTiles = nRowTiles * (HF / 16);
    if (wave >= nTiles) return;
    const int rt = wave >> 4;             // 16 column tiles
    const int ct = wave & 15;
    const int row0 = rt * 16, col0 = ct * 16;
    const int lane = threadIdx.x & 31;
    const int half = lane >> 4, l = lane & 15;
    const int arow = row0 + l;            // A: M = lane&15 (both halves)
    const int bcol = col0 + l;            // B/D: N = lane&15

    v8f acc = {};
    #pragma unroll
    for (int k = 0; k < F_DIM; k += 4) {
        const int ka = k + 2 * half;      // lanes 0-15: K,K+1 ; 16-31: K+2,K+3
        v2f a, b;
        a.x = h[arow * F_DIM + ka];
        a.y = h[arow * F_DIM + ka + 1];
        b.x = w[ka * HF + bcol];
        b.y = w[(ka + 1) * HF + bcol];
        acc = __builtin_amdgcn_wmma_f32_16x16x4_f32(
                  false, a, false, b, (short)0, acc, false, false);
    }
    const int orow = row0 + half * 8;     // D: lanes 0-15 rows 0-7, 16-31 rows 8-15
    #pragma unroll
    for (int i = 0; i < 8; ++i)
        feat[(size_t)(orow + i) * HF + bcol] = acc[i];
}

// ---------------------------------------------------------------------------
// 2) el[n,h] = <feat[n,h,:], attn_l[h,:]> ; er likewise. One wave per node.
//    attn_l flattened [H*F] matches feat's inner [H,F] layout.
// ---------------------------------------------------------------------------
__global__ void attn_score_kernel(const float* __restrict__ feat,
                                  const float* __restrict__ al,
                                  const float* __restrict__ ar,
                                  float* __restrict__ el, float* __restrict__ er,
                                  int N) {
    const int wave = (blockIdx.x * blockDim.x + threadIdx.x) >> 5;
    if (wave >= N) return;
    const int lane = threadIdx.x & 31;
    const int abase = lane * 8;           // lanes 0-7 -> head 0, 8-15 -> head 1, ...
    const size_t base = (size_t)wave * HF + abase;
    float sl = 0.0f, sr = 0.0f;
    #pragma unroll
    for (int j = 0; j < 8; ++j) {
        const float f = feat[base + j];
        sl += f * al[abase + j];
        sr += f * ar[abase + j];
    }
    #pragma unroll
    for (int msk = 1; msk < 8; msk <<= 1) {
        sl += __shfl_xor(sl, msk, 32);
        sr += __shfl_xor(sr, msk, 32);
    }
    if ((lane & 7) == 0) {
        const int hd = lane >> 3;
        el[wave * H_DIM + hd] = sl;
        er[wave * H_DIM + hd] = sr;
    }
}

// ---------------------------------------------------------------------------
// 3) segment max over incoming edges:  m[d,h] = max leaky_relu(el[s]+er[d])
// ---------------------------------------------------------------------------
__global__ void edge_max_kernel(const int* __restrict__ src, const int* __restrict__ dst,
                                const float* __restrict__ el, const float* __restrict__ er,
                                float* __restrict__ m, int EH) {
    int t = blockIdx.x * blockDim.x + threadIdx.x;
    if (t >= EH) return;
    const int e = t >> 2, hd = t & 3;
    const int s = src[e], d = dst[e];
    float v = el[s * H_DIM + hd] + er[d * H_DIM + hd];
    v = v > 0.0f ? v : NEG_SLOPE * v;
    atomic_max_f32(&m[d * H_DIM + hd], v);
}

// ---------------------------------------------------------------------------
// 4) ex = exp(e - m[dst]) ; z[d,h] += ex ; stash ex
// ---------------------------------------------------------------------------
__global__ void edge_exp_kernel(const int* __restrict__ src, const int* __restrict__ dst,
                                const float* __restrict__ el, const float* __restrict__ er,
                                const float* __restrict__ m, float* __restrict__ z,
                                float* __restrict__ exb, int EH) {
    int t = blockIdx.x * blockDim.x + threadIdx.x;
    if (t >= EH) return;
    const int e = t >> 2, hd = t & 3;
    const int s = src[e], d = dst[e];
    float v = el[s * H_DIM + hd] + er[d * H_DIM + hd];
    v = v > 0.0f ? v : NEG_SLOPE * v;
    const float ex = __expf(v - m[d * H_DIM + hd]);
    exb[t] = ex;
    atomicAdd(&z[d * H_DIM + hd], ex);
}

// ---------------------------------------------------------------------------
// 5) rst[dst,:] += feat[src,:] * alpha   — one wave per edge, 8 floats/lane.
//    rst is 51 MB -> resident in the 192 MB L2, atomics stay on-chip.
// ---------------------------------------------------------------------------
__global__ void aggregate_kernel(const int* __restrict__ src, const int* __restrict__ dst,
                                 const float* __restrict__ feat,
                                 const float* __restrict__ exb, const float* __restrict__ z,
                                 float* __restrict__ rst, int E) {
    const int e = (blockIdx.x * blockDim.x + threadIdx.x) >> 5;
    if (e >= E) return;
    const int lane = threadIdx.x & 31;
    const int s = src[e], d = dst[e];
    const int hd = lane >> 3;
    const float alpha = exb[e * H_DIM + hd] / z[d * H_DIM + hd];
    const int base = lane * 8;
    const float4* f4 = (const float4*)(feat + (size_t)s * HF + base);
    const float4 v0 = f4[0], v1 = f4[1];
    float* r = rst + (size_t)d * HF + base;
    atomicAdd(r + 0, v0.x * alpha); atomicAdd(r + 1, v0.y * alpha);
    atomicAdd(r + 2, v0.z * alpha); atomicAdd(r + 3, v0.w * alpha);
    atomicAdd(r + 4, v1.x * alpha); atomicAdd(r + 5, v1.y * alpha);
    atomicAdd(r + 6, v1.z * alpha); atomicAdd(r + 7, v1.w * alpha);
}

// ---------------------------------------------------------------------------
// 6) x[N,64] = (rst + gat_bias)[N,256] @ out_w[256,64] + out_b   (f32 WMMA)
// ---------------------------------------------------------------------------
__global__ void out_gemm_kernel(const float* __restrict__ rst,
                                const float* __restrict__ gb,
                                const float* __restrict__ w,
                                const float* __restrict__ ob,
                                float* __restrict__ x, int nRowTiles) {
    const int wave   = (blockIdx.x * blockDim.x + threadIdx.x) >> 5;
    const int nTiles = nRowTiles * (F_DIM / 16);
    if (wave >= nTiles) return;
    const int rt = wave >> 2;             // 4 column tiles
    const int ct = wave & 3;
    const int row0 = rt * 16, col0 = ct * 16;
    const int lane = threadIdx.x & 31;
    const int half = lane >> 4, l = lane & 15;
    const int arow = row0 + l;
    const int bcol = col0 + l;

    v8f acc = {};
    #pragma unroll 8
    for (int k = 0; k < HF; k += 4) {
        const int ka = k + 2 * half;
        v2f a, b;
        a.x = rst[(size_t)arow * HF + ka]     + gb[ka];
        a.y = rst[(size_t)arow * HF + ka + 1] + gb[ka + 1];
        b.x = w[ka * F_DIM + bcol];
        b.y = w[(ka + 1) * F_DIM + bcol];
        acc = __builtin_amdgcn_wmma_f32_16x16x4_f32(
                  false, a, false, b, (short)0, acc, false, false);
    }
    const float bias = ob[bcol];
    const int orow = row0 + half * 8;
    #pragma unroll
    for (int i = 0; i < 8; ++i)
        x[(size_t)(orow + i) * F_DIM + bcol] = acc[i] + bias;
}

// ---------------------------------------------------------------------------
// 7) layernorm over F=64 per node — one wave per node, 2 elems/lane
// ---------------------------------------------------------------------------
__global__ void layernorm_kernel(const float* __restrict__ x,
                                 const float* __restrict__ g,
                                 const float* __restrict__ b,
                                 float* __restrict__ out, int N) {
    const int wave = (blockIdx.x * blockDim.x + threadIdx.x) >> 5;
    if (wave >= N) return;
    const int lane = threadIdx.x & 31;
    const size_t base = (size_t)wave * F_DIM;
    const float x0 = x[base + lane], x1 = x[base + lane + 32];
    float s = x0 + x1, sq = x0 * x0 + x1 * x1;
    #pragma unroll
    for (int msk = 1; msk < 32; msk <<= 1) {
        s  += __shfl_xor(s,  msk, 32);
        sq += __shfl_xor(sq, msk, 32);
    }
    const float mu  = s * (1.0f / 64.0f);
    const float var = sq * (1.0f / 64.0f) - mu * mu;
    const float inv = rsqrtf(var + LN_EPS);
    out[base + lane]      = (x0 - mu) * inv * g[lane]      + b[lane];
    out[base + lane + 32] = (x1 - mu) * inv * g[lane + 32] + b[lane + 32];
}

// ---------------------------------------------------------------------------
extern "C" void kernel_launch(void* const* d_in, const int* in_sizes, int n_in,
                              void* d_out, int out_size, void* d_ws, size_t ws_size,
                              hipStream_t stream) {
    const float* h   = (const float*)d_in[0];
    const int*   src = (const int*)  d_in[1];
    const int*   dst = (const int*)  d_in[2];
    const float* fcw = (const float*)d_in[3];
    const float* al  = (const float*)d_in[4];
    const float* ar  = (const float*)d_in[5];
    const float* gb  = (const float*)d_in[6];
    const float* ow  = (const float*)d_in[7];
    const float* ob  = (const float*)d_in[8];
    const float* lng = (const float*)d_in[9];
    const float* lnb = (const float*)d_in[10];

    const int N = in_sizes[0] / F_DIM;   // 50000
    const int E = in_sizes[1];           // 800000

    // workspace carve-out (256B aligned slices)
    char*  ws  = (char*)d_ws;
    size_t off = 0;
    auto carve = [&](size_t bytes) -> float* {
        float* p = (float*)(ws + off);
        off += (bytes + 255) & ~(size_t)255;
        return p;
    };
    float* feat = carve((size_t)N * HF * 4);      // [N,256]
    float* rst  = carve((size_t)N * HF * 4);      // [N,256]
    float* el   = carve((size_t)N * H_DIM * 4);
    float* er   = carve((size_t)N * H_DIM * 4);
    float* m    = carve((size_t)N * H_DIM * 4);
    float* z    = carve((size_t)N * H_DIM * 4);
    float* exb  = carve((size_t)E * H_DIM * 4);   // [E,4]
    float* x    = carve((size_t)N * F_DIM * 4);   // [N,64]
    (void)ws_size;

    const int nRowTiles = (N + 15) / 16;

    { // init rst/z/m
        const int total  = N * HF;
        const int blocks = (total + 255) / 256;
        init_kernel<<<blocks, 256, 0, stream>>>(rst, z, m, total, N * H_DIM);
    }
    { // FC GEMM (WMMA)
        const int waves  = nRowTiles * (HF / 16);
        const int blocks = (waves + 7) / 8;
        fc_gemm_kernel<<<blocks, 256, 0, stream>>>(h, fcw, feat, nRowTiles);
    }
    { // attention scores
        const int blocks = (N + 7) / 8;
        attn_score_kernel<<<blocks, 256, 0, stream>>>(feat, al, ar, el, er, N);
    }
    { // edge segment-max
        const int EH = E * H_DIM;
        const int blocks = (EH + 255) / 256;
        edge_max_kernel<<<blocks, 256, 0, stream>>>(src, dst, el, er, m, EH);
        edge_exp_kernel<<<blocks, 256, 0, stream>>>(src, dst, el, er, m, z, exb, EH);
    }
    { // edge aggregation (dominant, wave per edge)
        const int blocks = (E + 7) / 8;
        aggregate_kernel<<<blocks, 256, 0, stream>>>(src, dst, feat, exb, z, rst, E);
    }
    { // output GEMM (WMMA)
        const int waves  = nRowTiles * (F_DIM / 16);
        const int blocks = (waves + 7) / 8;
        out_gemm_kernel<<<blocks, 256, 0, stream>>>(rst, gb, ow, ob, x, nRowTiles);
    }
    { // layernorm -> d_out
        const int blocks = (N + 7) / 8;
        layernorm_kernel<<<blocks, 256, 0, stream>>>(x, lng, lnb, (float*)d_out, N);
    }
}